// ForceDirectedProcessor_6536940224657
// MI455X (gfx1250) — compile-verified
//
#include <hip/hip_runtime.h>

typedef float v2f __attribute__((ext_vector_type(2)));
typedef float v8f __attribute__((ext_vector_type(8)));

#define FEAT    128
#define TROWS   16
#define LSTRIDE 132   // 128 + 4 pad -> conflict-free column walks (132 % 64 == 4)

// ---------------------------------------------------------------------------
// Kernel 0: zero the segment-sum accumulator (agg half of d_out) and counts.
// ---------------------------------------------------------------------------
__global__ void fdp_zero(float* __restrict__ agg_acc, float* __restrict__ counts,
                         int n_agg, int n_cnt) {
  const int stride = gridDim.x * blockDim.x;
  for (int i = blockIdx.x * blockDim.x + threadIdx.x; i < n_agg; i += stride)
    agg_acc[i] = 0.0f;
  for (int i = blockIdx.x * blockDim.x + threadIdx.x; i < n_cnt; i += stride)
    counts[i] = 0.0f;
}

// ---------------------------------------------------------------------------
// Kernel 1: scatter-add. One wave per edge; lane L owns features [4L, 4L+4).
// Coalesced b128 loads of the message row; native f32 atomics into sums
// (51 MB surface -> L2-resident on the 192 MB L2).
// ---------------------------------------------------------------------------
__global__ void fdp_scatter(const float4* __restrict__ msg4, const int* __restrict__ col,
                            float* __restrict__ sums, float* __restrict__ counts,
                            int nedges) {
  const int tid  = blockIdx.x * blockDim.x + threadIdx.x;
  const int e    = tid >> 5;
  const int lane = tid & 31;
  if (e >= nedges) return;

  const int    c = col[e];
  const float4 v = msg4[(size_t)e * (FEAT / 4) + lane];
  float* dst = sums + (size_t)c * FEAT + lane * 4;

  __hip_atomic_fetch_add(dst + 0, v.x, __ATOMIC_RELAXED, __HIP_MEMORY_SCOPE_AGENT);
  __hip_atomic_fetch_add(dst + 1, v.y, __ATOMIC_RELAXED, __HIP_MEMORY_SCOPE_AGENT);
  __hip_atomic_fetch_add(dst + 2, v.z, __ATOMIC_RELAXED, __HIP_MEMORY_SCOPE_AGENT);
  __hip_atomic_fetch_add(dst + 3, v.w, __ATOMIC_RELAXED, __HIP_MEMORY_SCOPE_AGENT);
  if (lane == 0)
    __hip_atomic_fetch_add(counts + c, 1.0f, __ATOMIC_RELAXED, __HIP_MEMORY_SCOPE_AGENT);
}

// ---------------------------------------------------------------------------
// Kernel 2: per-node pipeline. One wave handles a 16-row tile.
// Row reductions (sum, sum-of-squares over F=128) are done with
// V_WMMA_F32_16X16X4_F32 against an all-ones B matrix, accumulating D->C
// over 32 K-steps. A-operand layout per ISA: lanes 0-15 hold (M=L, K=k0,k0+1),
// lanes 16-31 hold (M=L-16, K=k0+2,k0+3). C/D layout: lanes 0-15 carry rows
// 0-7 in c[0..7], lanes 16-31 carry rows 8-15.
// ---------------------------------------------------------------------------
__global__ void __launch_bounds__(32)
fdp_node(const float* __restrict__ x, float* __restrict__ agg_io,
         float* __restrict__ fx_out, const float* __restrict__ counts,
         const float* __restrict__ g1, const float* __restrict__ b1,
         const float* __restrict__ g2, const float* __restrict__ b2,
         const float* __restrict__ wrep, int N) {
  __shared__ float tile[TROWS * LSTRIDE];
  __shared__ float s_mu[TROWS];
  __shared__ float s_rs[TROWS];

  const int L     = threadIdx.x;
  const int r     = L & 15;
  const int khalf = (L >> 4) << 1;           // 0 for lanes 0-15, 2 for lanes 16-31
  const int base  = blockIdx.x * TROWS;
  const int row   = base + r;
  const int rowc  = row < N ? row : N - 1;   // clamp keeps EXEC all-ones for WMMA

  const float cnt = counts[rowc];
  const float inv = 1.0f / fmaxf(cnt, 1.0f);

  const float* __restrict__ srow = agg_io + (size_t)rowc * FEAT;
  const v2f ones = {1.0f, 1.0f};
  float* tlane = tile + r * LSTRIDE + khalf;

  // ---- Pass A: agg = sums * inv ; WMMA-accumulate rowsum & rowsumsq; stash in LDS
  v8f cs = {};
  v8f cq = {};
#pragma unroll
  for (int k0 = 0; k0 < FEAT; k0 += 4) {
    v2f a;
    a.x = srow[k0 + khalf] * inv;
    a.y = srow[k0 + khalf + 1] * inv;
    tlane[k0]     = a.x;
    tlane[k0 + 1] = a.y;
    v2f a2 = a * a;
    cs = __builtin_amdgcn_wmma_f32_16x16x4_f32(false, a,  false, ones, (short)0, cs, false, false);
    cq = __builtin_amdgcn_wmma_f32_16x16x4_f32(false, a2, false, ones, (short)0, cq, false, false);
  }
  if (L == 0 || L == 16) {
    const int mbase = (L >> 4) << 3;
#pragma unroll
    for (int v = 0; v < 8; ++v) {
      float mu  = cs[v] * (1.0f / FEAT);
      float var = cq[v] * (1.0f / FEAT) - mu * mu;
      s_mu[mbase + v] = mu;
      s_rs[mbase + v] = rsqrtf(var + 1e-5f);
    }
  }
  __syncthreads();

  // ---- Pass B: agg_ln -> global ; y = x + (x - agg_ln)*w -> back into LDS tile
  const float4 g1v = ((const float4*)g1)[L];
  const float4 b1v = ((const float4*)b1)[L];
  const float4 g2v = ((const float4*)g2)[L];
  const float4 b2v = ((const float4*)b2)[L];
  const float4 wv  = ((const float4*)wrep)[L];
  const int c0 = L << 2;

#pragma unroll 4
  for (int rr = 0; rr < TROWS; ++rr) {
    const int  orow = base + rr;
    const bool ok   = orow < N;
    const float mu = s_mu[rr];
    const float rs = s_rs[rr];
    float* tp = tile + rr * LSTRIDE + c0;
    const float4 a  = *(const float4*)tp;
    const float4 xa = ((const float4*)(x + (size_t)(ok ? orow : 0) * FEAT))[L];
    float4 agg, y;
    agg.x = (a.x - mu) * rs * g1v.x + b1v.x;
    agg.y = (a.y - mu) * rs * g1v.y + b1v.y;
    agg.z = (a.z - mu) * rs * g1v.z + b1v.z;
    agg.w = (a.w - mu) * rs * g1v.w + b1v.w;
    y.x = xa.x + (xa.x - agg.x) * wv.x;
    y.y = xa.y + (xa.y - agg.y) * wv.y;
    y.z = xa.z + (xa.z - agg.z) * wv.z;
    y.w = xa.w + (xa.w - agg.w) * wv.w;
    if (ok) ((float4*)(agg_io + (size_t)orow * FEAT))[L] = agg;
    *(float4*)tp = y;
  }
  __syncthreads();

  // ---- Pass C: WMMA-reduce y tile from LDS (same A layout, now ds_ reads)
  v8f ys = {};
  v8f yq = {};
#pragma unroll
  for (int k0 = 0; k0 < FEAT; k0 += 4) {
    v2f a;
    a.x = tlane[k0];
    a.y = tlane[k0 + 1];
    v2f a2 = a * a;
    ys = __builtin_amdgcn_wmma_f32_16x16x4_f32(false, a,  false, ones, (short)0, ys, false, false);
    yq = __builtin_amdgcn_wmma_f32_16x16x4_f32(false, a2, false, ones, (short)0, yq, false, false);
  }
  if (L == 0 || L == 16) {
    const int mbase = (L >> 4) << 3;
#pragma unroll
    for (int v = 0; v < 8; ++v) {
      float mu  = ys[v] * (1.0f / FEAT);
      float var = yq[v] * (1.0f / FEAT) - mu * mu;
      s_mu[mbase + v] = mu;
      s_rs[mbase + v] = rsqrtf(var + 1e-5f);
    }
  }
  __syncthreads();

  // ---- Pass D: fx = LN2(y) -> global
#pragma unroll 4
  for (int rr = 0; rr < TROWS; ++rr) {
    const int orow = base + rr;
    if (orow >= N) break;
    const float mu = s_mu[rr];
    const float rs = s_rs[rr];
    const float4 y = *(const float4*)(tile + rr * LSTRIDE + c0);
    float4 fx;
    fx.x = (y.x - mu) * rs * g2v.x + b2v.x;
    fx.y = (y.y - mu) * rs * g2v.y + b2v.y;
    fx.z = (y.z - mu) * rs * g2v.z + b2v.z;
    fx.w = (y.w - mu) * rs * g2v.w + b2v.w;
    ((float4*)(fx_out + (size_t)orow * FEAT))[L] = fx;
  }
}

// ---------------------------------------------------------------------------
extern "C" void kernel_launch(void* const* d_in, const int* in_sizes, int n_in,
                              void* d_out, int out_size, void* d_ws, size_t ws_size,
                              hipStream_t stream) {
  const float* x   = (const float*)d_in[0];   // [N, 128]
  const float* msg = (const float*)d_in[1];   // [E, 128]
  const float* g1  = (const float*)d_in[2];
  const float* b1  = (const float*)d_in[3];
  const float* g2  = (const float*)d_in[4];
  const float* b2  = (const float*)d_in[5];
  const float* w   = (const float*)d_in[6];
  const int*   col = (const int*)d_in[7];     // harness: integer -> const int*

  const int N = in_sizes[0] / FEAT;
  const int E = in_sizes[1] / FEAT;

  float* fx  = (float*)d_out;                       // output 0: [N,128]
  float* agg = (float*)d_out + (size_t)N * FEAT;    // output 1: [N,128]; doubles as sum accumulator
  float* counts = (float*)d_ws;                     // N floats of scratch

  {
    const int n_agg = N * FEAT;
    int blocks = (n_agg + 255) / 256;
    if (blocks > 4096) blocks = 4096;
    fdp_zero<<<blocks, 256, 0, stream>>>(agg, counts, n_agg, N);
  }
  {
    const long long threads = (long long)E * 32;
    const int blocks = (int)((threads + 255) / 256);
    fdp_scatter<<<blocks, 256, 0, stream>>>((const float4*)msg, col, agg, counts, E);
  }
  {
    const int blocks = (N + TROWS - 1) / TROWS;
    fdp_node<<<blocks, 32, 0, stream>>>(x, agg, fx, counts, g1, b1, g2, b2, w, N);
  }
}